// VQ_Payam_8821862826423
// MI455X (gfx1250) — compile-verified
//
#include <hip/hip_runtime.h>
#include <hip/hip_bf16.h>

typedef __attribute__((ext_vector_type(2))) float v2f;
typedef __attribute__((ext_vector_type(8))) float v8f;

#define NN 32768      // number of input vectors
#define DD 256        // embedding dim
#define KK 1024       // codebook size
#define XPAD 258      // padded LDS row stride (floats) -> conflict-free A reads

// ---------------------------------------------------------------------------
// wnorm[k] = sum_d w[k][d]^2 ; also zero histogram counts
// one wave per codebook row, 8 rows per 256-thread block
// ---------------------------------------------------------------------------
__global__ __launch_bounds__(256) void k_wnorm(const float* __restrict__ w,
                                               float* __restrict__ wnorm,
                                               int* __restrict__ counts) {
    int row  = blockIdx.x * 8 + (threadIdx.x >> 5);
    int lane = threadIdx.x & 31;
    const float* r = w + (long)row * DD;
    float s = 0.f;
    #pragma unroll
    for (int d = lane; d < DD; d += 32) { float t = r[d]; s += t * t; }
    #pragma unroll
    for (int m = 16; m >= 1; m >>= 1) s += __shfl_xor(s, m, 32);
    if (lane == 0) { wnorm[row] = s; counts[row] = 0; }
}

// ---------------------------------------------------------------------------
// zero the encodings region (float2 stores; region is 8-byte aligned)
// ---------------------------------------------------------------------------
__global__ __launch_bounds__(256) void k_zero_enc(float2* __restrict__ p, long n2) {
    long i      = (long)blockIdx.x * blockDim.x + threadIdx.x;
    long stride = (long)gridDim.x * blockDim.x;
    float2 z; z.x = 0.f; z.y = 0.f;
    for (; i < n2; i += stride) p[i] = z;
}

// ---------------------------------------------------------------------------
// argmin kernel: one block = 16 rows of x vs all 1024 codes, fp32 WMMA GEMM
// dist(n,k) = ||w_k||^2 - 2 * <x_n, w_k>   (||x||^2 dropped: row-constant)
// ---------------------------------------------------------------------------
__global__ __launch_bounds__(256) void k_argmin(const float* __restrict__ x,
                                                const float* __restrict__ w,
                                                const float* __restrict__ wnorm,
                                                int* __restrict__ idx_out) {
    __shared__ float xs[16 * XPAD];
    __shared__ float red_val[8][16];
    __shared__ int   red_idx[8][16];

    const int m0  = blockIdx.x * 16;
    const int tid = threadIdx.x;

    // stage 16 rows of x into LDS (padded stride)
    for (int i = tid; i < 16 * DD; i += 256) {
        int r = i >> 8;
        int c = i & (DD - 1);
        xs[r * XPAD + c] = x[(long)(m0 + r) * DD + c];
    }
    __syncthreads();

    const int wave = tid >> 5;
    const int lane = tid & 31;
    const int lrow = lane & 15;             // A: row M ; B: col N within tile
    const int koff = (lane < 16) ? 0 : 2;   // K sub-offset per ISA layout

    float minv[8];
    int   mini[8];
    #pragma unroll
    for (int v = 0; v < 8; ++v) { minv[v] = 3.402823e38f; mini[v] = 0; }

    const float* xrow = xs + lrow * XPAD + koff;

    for (int t = 0; t < 8; ++t) {
        const int c0 = (wave + t * 8) * 16;            // column-tile base
        const float* wrow = w + (long)(c0 + lrow) * DD + koff;

        v8f acc = {};
        #pragma unroll 8
        for (int d = 0; d < DD; d += 4) {
            v2f a; a.x = xrow[d]; a.y = xrow[d + 1];   // A: 16x4 fp32 fragment
            v2f b; b.x = wrow[d]; b.y = wrow[d + 1];   // B: 4x16 fp32 fragment
            acc = __builtin_amdgcn_wmma_f32_16x16x4_f32(
                false, a, false, b, (short)0, acc, false, false);
        }

        // acc[v]: lane<16 -> (M=v, N=c0+lane) ; lane>=16 -> (M=v+8, N=c0+lane-16)
        float wn = wnorm[c0 + lrow];
        #pragma unroll
        for (int v = 0; v < 8; ++v) {
            float dist = wn - 2.0f * acc[v];
            int   col  = c0 + lrow;
            if (dist < minv[v] || (dist == minv[v] && col < mini[v])) {
                minv[v] = dist; mini[v] = col;
            }
        }
    }

    // reduce across the 16 lanes of each half (rows stay within a half)
    #pragma unroll
    for (int m = 8; m >= 1; m >>= 1) {
        #pragma unroll
        for (int v = 0; v < 8; ++v) {
            float ov = __shfl_xor(minv[v], m, 32);
            int   oi = __shfl_xor(mini[v], m, 32);
            if (ov < minv[v] || (ov == minv[v] && oi < mini[v])) {
                minv[v] = ov; mini[v] = oi;
            }
        }
    }
    if (lane == 0 || lane == 16) {
        int mbase = (lane == 0) ? 0 : 8;
        #pragma unroll
        for (int v = 0; v < 8; ++v) {
            red_val[wave][mbase + v] = minv[v];
            red_idx[wave][mbase + v] = mini[v];
        }
    }
    __syncthreads();

    // reduce across the 8 waves (each covered disjoint column tiles)
    if (tid < 16) {
        float bv = red_val[0][tid];
        int   bi = red_idx[0][tid];
        #pragma unroll
        for (int wv = 1; wv < 8; ++wv) {
            float ov = red_val[wv][tid];
            int   oi = red_idx[wv][tid];
            if (ov < bv || (ov == bv && oi < bi)) { bv = ov; bi = oi; }
        }
        idx_out[m0 + tid] = bi;
    }
}

// ---------------------------------------------------------------------------
// gather quantized rows, write them out, accumulate squared error per block
// grid must be exactly 1024 blocks (partials array size)
// ---------------------------------------------------------------------------
__global__ __launch_bounds__(256) void k_quant(const float* __restrict__ x,
                                               const float* __restrict__ w,
                                               const int* __restrict__ idx,
                                               float* __restrict__ q_out,
                                               float* __restrict__ partial) {
    __shared__ float sred[256];
    const long NT = (long)NN * DD;
    long i      = (long)blockIdx.x * 256 + threadIdx.x;
    long stride = (long)gridDim.x * 256;
    float acc = 0.f;
    for (; i < NT; i += stride) {
        int n = (int)(i >> 8);
        int d = (int)(i & (DD - 1));
        float qv = w[(long)idx[n] * DD + d];
        float xv = x[i];
        q_out[i] = qv;                 // quantized_st == quantized numerically
        float df = qv - xv;
        acc += df * df;
    }
    sred[threadIdx.x] = acc;
    __syncthreads();
    #pragma unroll
    for (int s = 128; s >= 1; s >>= 1) {
        if (threadIdx.x < s) sred[threadIdx.x] += sred[threadIdx.x + s];
        __syncthreads();
    }
    if (threadIdx.x == 0) partial[blockIdx.x] = sred[0];
}

// ---------------------------------------------------------------------------
// one-hot scatter + integer histogram
// ---------------------------------------------------------------------------
__global__ __launch_bounds__(256) void k_scatter(const int* __restrict__ idx,
                                                 float* __restrict__ enc,
                                                 int* __restrict__ counts) {
    int n = blockIdx.x * blockDim.x + threadIdx.x;   // grid covers N exactly
    int k = idx[n];
    enc[(long)n * KK + k] = 1.0f;
    atomicAdd(&counts[k], 1);
}

// ---------------------------------------------------------------------------
// final: loss = 1.25 * sum(partials)/(N*D) ; perplexity = exp(-sum p log(p+eps))
// ---------------------------------------------------------------------------
__global__ __launch_bounds__(1024) void k_final(const float* __restrict__ partial,
                                                const int* __restrict__ counts,
                                                float* __restrict__ loss_out,
                                                float* __restrict__ perp_out) {
    __shared__ float s1[1024];
    __shared__ float s2[1024];
    int t = threadIdx.x;
    s1[t] = partial[t];
    float p = (float)counts[t] / (float)NN;
    s2[t] = -p * logf(p + 1e-10f);
    __syncthreads();
    #pragma unroll
    for (int s = 512; s >= 1; s >>= 1) {
        if (t < s) { s1[t] += s1[t + s]; s2[t] += s2[t + s]; }
        __syncthreads();
    }
    if (t == 0) {
        loss_out[0] = 1.25f * s1[0] / (float)((long)NN * DD);
        perp_out[0] = expf(s2[0]);
    }
}

// ---------------------------------------------------------------------------
extern "C" void kernel_launch(void* const* d_in, const int* in_sizes, int n_in,
                              void* d_out, int out_size, void* d_ws, size_t ws_size,
                              hipStream_t stream) {
    const float* x = (const float*)d_in[0];   // [16,2048,256] fp32
    const float* w = (const float*)d_in[1];   // [1024,256]    fp32

    float* out = (float*)d_out;
    // tuple layout: loss[1] | quantized_st[N*D] | perplexity[1] | encodings[N*K]
    float* loss_out = out;
    float* q_out    = out + 1;
    float* perp_out = out + 1 + (long)NN * DD;
    float* enc_out  = out + 2 + (long)NN * DD;

    // workspace layout
    char*  ws      = (char*)d_ws;
    int*   idx     = (int*)(ws);                         // 32768 ints
    float* wnorm   = (float*)(ws + 32768 * 4);           // 1024 floats
    int*   counts  = (int*)(ws + 32768 * 4 + 4096);      // 1024 ints
    float* partial = (float*)(ws + 32768 * 4 + 8192);    // 1024 floats

    k_wnorm  <<<KK / 8, 256, 0, stream>>>(w, wnorm, counts);
    k_zero_enc<<<4096, 256, 0, stream>>>((float2*)enc_out, (long)NN * KK / 2);
    k_argmin <<<NN / 16, 256, 0, stream>>>(x, w, wnorm, idx);
    k_quant  <<<1024, 256, 0, stream>>>(x, w, idx, q_out, partial);
    k_scatter<<<NN / 256, 256, 0, stream>>>(idx, enc_out, counts);
    k_final  <<<1, 1024, 0, stream>>>(partial, counts, loss_out, perp_out);
}